// Sparse_MoE_38912403702038
// MI455X (gfx1250) — compile-verified
//
#include <hip/hip_runtime.h>
#include <hip/hip_bf16.h>

#define D_MODEL 1024
#define NEXP 8
#define MB 64          // token-tile rows per block
#define NB 64          // output-column tile per block
#define KB 32          // k-chunk staged in LDS
#define KTILES (D_MODEL / KB)
#define LDS_STRIDE (KB + 4)   // 36 floats = 144B row stride; 144 % 16 == 0

typedef __attribute__((ext_vector_type(2))) float v2f;
typedef __attribute__((ext_vector_type(8))) float v8f;

// ---------------------------------------------------------------------------
// Kernel 0: zero the output accumulator and per-expert counters.
// ---------------------------------------------------------------------------
__global__ __launch_bounds__(256) void moe_zero(float* __restrict__ out,
                                                int* __restrict__ counts,
                                                size_t total) {
  size_t i = (size_t)blockIdx.x * blockDim.x + threadIdx.x;
  if (i < total) out[i] = 0.0f;
  if (blockIdx.x == 0 && threadIdx.x < NEXP) counts[threadIdx.x] = 0;
}

// ---------------------------------------------------------------------------
// Kernel 1: gating. One block (8 waves) per token; wave w computes logit for
// expert w via lane-strided dot + wave32 shuffle reduction. Thread 0 picks
// top-2 (lower index wins ties, matching lax.top_k), softmaxes, and appends
// (token, weight) to each chosen expert's list.
// ---------------------------------------------------------------------------
__global__ __launch_bounds__(256) void moe_gate(const float* __restrict__ x,
                                                const float* __restrict__ Wg,
                                                const float* __restrict__ bg,
                                                int* __restrict__ counts,
                                                int* __restrict__ tok_ids,
                                                float* __restrict__ tok_w,
                                                int N) {
  const int n    = blockIdx.x;
  const int lane = threadIdx.x & 31;
  const int wave = threadIdx.x >> 5;      // 0..7 == expert id

  const float* xr = x  + (size_t)n    * D_MODEL;
  const float* wr = Wg + (size_t)wave * D_MODEL;

  float s = 0.0f;
  for (int i = lane; i < D_MODEL; i += 32) s += xr[i] * wr[i];
#pragma unroll
  for (int off = 16; off >= 1; off >>= 1) s += __shfl_xor(s, off, 32);

  __shared__ float logits[NEXP];
  if (lane == 0) logits[wave] = s + bg[wave];
  __syncthreads();

  if (threadIdx.x == 0) {
    int b0 = 0; float v0 = logits[0];
#pragma unroll
    for (int e = 1; e < NEXP; e++)
      if (logits[e] > v0) { v0 = logits[e]; b0 = e; }
    int b1 = -1; float v1 = -3.4e38f;
#pragma unroll
    for (int e = 0; e < NEXP; e++)
      if (e != b0 && logits[e] > v1) { v1 = logits[e]; b1 = e; }
    // softmax over [v0, v1] with v0 >= v1
    float t  = expf(v1 - v0);
    float w0 = 1.0f / (1.0f + t);
    float w1 = t * w0;
    int p0 = atomicAdd(&counts[b0], 1);
    tok_ids[(size_t)b0 * N + p0] = n;
    tok_w  [(size_t)b0 * N + p0] = w0;
    int p1 = atomicAdd(&counts[b1], 1);
    tok_ids[(size_t)b1 * N + p1] = n;
    tok_w  [(size_t)b1 * N + p1] = w1;
  }
}

// ---------------------------------------------------------------------------
// Kernel 2: gathered expert GEMM using V_WMMA_F32_16X16X4_F32 with
// double-buffered GLOBAL_LOAD_ASYNC_TO_LDS_B128 staging (ASYNCcnt pipeline).
// Block = 256 threads (8 waves) computes a 64-token x 64-column tile for one
// expert. Padded rows (tid < 0) simply re-read token row 0: junk in A-row m
// only affects D-row m, which is never stored.
// ---------------------------------------------------------------------------
__global__ __launch_bounds__(256) void moe_expert_gemm(
    const float* __restrict__ x, const float* __restrict__ We,
    const float* __restrict__ be, const int* __restrict__ counts,
    const int* __restrict__ tok_ids, const float* __restrict__ tok_w,
    float* __restrict__ out, int N) {
  const int e   = blockIdx.z;
  const int cnt = counts[e];
  const int m0  = blockIdx.x * MB;
  if (m0 >= cnt) return;                 // uniform early-exit (EXEC stays full)
  const int valid = min(MB, cnt - m0);
  const int j0    = blockIdx.y * NB;

  __shared__ float Xs[2][MB][LDS_STRIDE];
  __shared__ float Ws[2][NB][LDS_STRIDE];
  __shared__ int   tidS[MB];
  __shared__ float wS[MB];

  const int t = threadIdx.x;
  if (t < MB) {
    const int tid = (t < valid) ? tok_ids[(size_t)e * N + m0 + t] : -1;
    tidS[t] = tid;
    wS[t]   = (t < valid) ? tok_w[(size_t)e * N + m0 + t] : 0.0f;
  }
  __syncthreads();

  const int lane = t & 31;
  const int wave = t >> 5;
  const int wm   = wave & 3;            // 16-row sub-tile
  const int wn   = wave >> 2;           // 32-col sub-tile
  const int lr   = lane & 15;           // row/col within 16
  const int lh   = lane >> 4;           // lane half -> k = 2*lh + v

  // staging assignment: 4 threads per row, each moves 8 consecutive floats
  const int lrow = t >> 2;              // 0..63
  const int lcol = (t & 3) * 8;         // 0,8,16,24 (32B-aligned chunks)
  const int rtid = tidS[lrow];
  const int xrow = (rtid >= 0) ? rtid : 0;   // padded rows: harmless duplicate
  const float* xsrc = x + (size_t)xrow * D_MODEL + lcol;
  const float* wsrc = We + (size_t)e * D_MODEL * D_MODEL +
                      (size_t)(j0 + lrow) * D_MODEL + lcol;

  // Async prefetch of one (X,W) k-chunk into LDS buffer `buf`.
  // inst OFFSET is added to BOTH the LDS and global addresses (ISA 10.7/15.18),
  // so one LDS-address VGPR covers both 16B halves.
  auto prefetch = [&](int buf, int k0) {
    unsigned xl = (unsigned)(uintptr_t)&Xs[buf][lrow][lcol];
    unsigned wl = (unsigned)(uintptr_t)&Ws[buf][lrow][lcol];
    unsigned long long xg = (unsigned long long)(uintptr_t)(xsrc + k0);
    unsigned long long wg = (unsigned long long)(uintptr_t)(wsrc + k0);
    asm volatile(
        "global_load_async_to_lds_b128 %0, %2, off\n\t"
        "global_load_async_to_lds_b128 %0, %2, off offset:16\n\t"
        "global_load_async_to_lds_b128 %1, %3, off\n\t"
        "global_load_async_to_lds_b128 %1, %3, off offset:16"
        :
        : "v"(xl), "v"(wl), "v"(xg), "v"(wg)
        : "memory");
  };

  v8f acc0 = {};
  v8f acc1 = {};

  prefetch(0, 0);                        // prologue: 4 async ops in flight

  for (int kt = 0; kt < KTILES; ++kt) {
    const int buf = kt & 1;
    if (kt + 1 < KTILES) {
      prefetch(buf ^ 1, (kt + 1) * KB);  // 8 in flight
      // ASYNCcnt load-completions retire in order: <=4 left => current tile done
      asm volatile("s_wait_asynccnt 0x4" ::: "memory");
    } else {
      asm volatile("s_wait_asynccnt 0x0" ::: "memory");
    }
    __syncthreads();                     // all waves' tile data visible

#pragma unroll
    for (int kk = 0; kk < KB; kk += 4) {
      // A fragment: row = lane%16 (token), k = kk + 2*lh + {0,1}
      v2f a  = *(const v2f*)&Xs[buf][wm * 16 + lr][kk + 2 * lh];
      // B fragments: col = lane%16 (output j), same k striping
      v2f bA = *(const v2f*)&Ws[buf][wn * 32 + lr][kk + 2 * lh];
      v2f bB = *(const v2f*)&Ws[buf][wn * 32 + 16 + lr][kk + 2 * lh];
      acc0 = __builtin_amdgcn_wmma_f32_16x16x4_f32(
          false, a, false, bA, (short)0, acc0, false, false);
      acc1 = __builtin_amdgcn_wmma_f32_16x16x4_f32(
          false, a, false, bB, (short)0, acc1, false, false);
    }
    __syncthreads();                     // done reading `buf`; safe to overwrite
  }

  // Epilogue: C/D layout — VGPR r holds M = r + 8*(lane/16), N = lane%16.
  const int jA = j0 + wn * 32 + lr;
  const int jB = jA + 16;
  const float biasA = be[(size_t)e * D_MODEL + jA];
  const float biasB = be[(size_t)e * D_MODEL + jB];
#pragma unroll
  for (int r = 0; r < 8; r++) {
    const int ml  = wm * 16 + r + 8 * lh;
    const int tid = tidS[ml];
    if (tid >= 0) {
      const float gw = wS[ml];
      atomicAdd(&out[(size_t)tid * D_MODEL + jA], gw * (acc0[r] + biasA));
      atomicAdd(&out[(size_t)tid * D_MODEL + jB], gw * (acc1[r] + biasB));
    }
  }
}

// ---------------------------------------------------------------------------
// Launcher
// ---------------------------------------------------------------------------
extern "C" void kernel_launch(void* const* d_in, const int* in_sizes, int n_in,
                              void* d_out, int out_size, void* d_ws, size_t ws_size,
                              hipStream_t stream) {
  const float* x  = (const float*)d_in[0];
  const float* We = (const float*)d_in[1];
  const float* be = (const float*)d_in[2];
  const float* Wg = (const float*)d_in[3];
  const float* bg = (const float*)d_in[4];
  float* out = (float*)d_out;

  const int N = in_sizes[0] / D_MODEL;   // 8192 tokens

  // Workspace layout: [counts: 8 ints][pad to 256B][tok_ids: 8*N int][tok_w: 8*N f32]
  char* ws = (char*)d_ws;
  int*   counts  = (int*)ws;
  int*   tok_ids = (int*)(ws + 256);
  float* tok_w   = (float*)(ws + 256 + (size_t)NEXP * N * sizeof(int));

  const size_t total = (size_t)N * D_MODEL;
  moe_zero<<<(unsigned)((total + 255) / 256), 256, 0, stream>>>(out, counts, total);

  moe_gate<<<N, 256, 0, stream>>>(x, Wg, bg, counts, tok_ids, tok_w, N);

  dim3 grid((N + MB - 1) / MB, D_MODEL / NB, NEXP);
  moe_expert_gemm<<<grid, 256, 0, stream>>>(x, We, be, counts, tok_ids, tok_w,
                                            out, N);
}